// SAModule_20160576487542
// MI455X (gfx1250) — compile-verified
//
#include <hip/hip_runtime.h>
#include <stdint.h>

#define B_    16
#define N_    4096
#define S_    1024
#define K_    32
#define C_    128
#define CIN_  131
#define P_    (B_ * S_ * K_)   /* 524288 columns */
#define BS_   (B_ * S_)        /* 16384 */
#define NBLK_ (P_ / 64)        /* 8192 GEMM blocks */
#define R2_   0.36f
#define EPS_  1e-5f

typedef float v2f __attribute__((ext_vector_type(2)));
typedef float v8f __attribute__((ext_vector_type(8)));

// ---------------------------------------------------------------------------
// Farthest point sampling: one block per batch, distance array in LDS.
// ---------------------------------------------------------------------------
__global__ __launch_bounds__(256)
void fps_kernel(const float* __restrict__ xyz, int* __restrict__ fps) {
  __shared__ float dist[N_];
  __shared__ float rd[256];
  __shared__ int   ri[256];
  __shared__ int   sFar;
  const int b = blockIdx.x;
  const int t = threadIdx.x;
  const float* xb = xyz + (size_t)b * N_ * 3;
  for (int n = t; n < N_; n += 256) dist[n] = 1e10f;
  if (t == 0) sFar = 0;
  __syncthreads();
  for (int j = 0; j < S_; ++j) {
    const int far = sFar;
    if (t == 0) fps[b * S_ + j] = far;
    const float cx = xb[far * 3 + 0];
    const float cy = xb[far * 3 + 1];
    const float cz = xb[far * 3 + 2];
    float bd = -1.0f; int bi = 0;
    for (int n = t; n < N_; n += 256) {
      float dx = xb[n * 3 + 0] - cx;
      float dy = xb[n * 3 + 1] - cy;
      float dz = xb[n * 3 + 2] - cz;
      float d  = dx * dx + dy * dy + dz * dz;
      float nd = fminf(dist[n], d);
      dist[n] = nd;
      if (nd > bd) { bd = nd; bi = n; }
    }
    rd[t] = bd; ri[t] = bi;
    __syncthreads();
    for (int sft = 128; sft > 0; sft >>= 1) {
      if (t < sft) {
        float od = rd[t + sft]; int oi = ri[t + sft];
        if (od > rd[t] || (od == rd[t] && oi < ri[t])) { rd[t] = od; ri[t] = oi; }
      }
      __syncthreads();
    }
    if (t == 0) sFar = ri[0];
    __syncthreads();
  }
}

// ---------------------------------------------------------------------------
// Gather new_xyz (written straight into d_out region 0).
// ---------------------------------------------------------------------------
__global__ void gather_newxyz(const float* __restrict__ xyz,
                              const int* __restrict__ fps,
                              float* __restrict__ out) {
  int t = blockIdx.x * 256 + threadIdx.x;
  if (t >= BS_) return;
  int b = t / S_;
  int n = fps[t];
  const float* p = xyz + ((size_t)b * N_ + n) * 3;
  out[t * 3 + 0] = p[0];
  out[t * 3 + 1] = p[1];
  out[t * 3 + 2] = p[2];
}

// ---------------------------------------------------------------------------
// Ball query: one wave32 per center; ordered compaction via ballot+popcount.
// Picks the 32 lowest indices with d^2 <= r^2 (matches the sort trick).
// ---------------------------------------------------------------------------
__global__ __launch_bounds__(128)
void ball_query_kernel(const float* __restrict__ xyz,
                       const float* __restrict__ nxyz,
                       int* __restrict__ idx) {
  const int lane = threadIdx.x & 31;
  const int bs   = blockIdx.x * 4 + (threadIdx.x >> 5);
  const int b    = bs >> 10;
  const float cx = nxyz[bs * 3 + 0];
  const float cy = nxyz[bs * 3 + 1];
  const float cz = nxyz[bs * 3 + 2];
  const float* xb = xyz + (size_t)b * N_ * 3;
  int* out = idx + (size_t)bs * K_;
  int cnt = 0;
  for (int n0 = 0; n0 < N_; n0 += 32) {
    int n = n0 + lane;
    float dx = xb[n * 3 + 0] - cx;
    float dy = xb[n * 3 + 1] - cy;
    float dz = xb[n * 3 + 2] - cz;
    bool in = (dx * dx + dy * dy + dz * dz) <= R2_;
    unsigned m = __builtin_amdgcn_ballot_w32(in);
    int slot = cnt + __builtin_popcount(m & ((1u << lane) - 1u));
    if (in && slot < K_) out[slot] = n;
    cnt += __builtin_popcount(m);
    if (cnt >= K_) break;
  }
  if (cnt < K_) {
    int first = (cnt > 0) ? out[0] : (N_ - 1);
    for (int tt = cnt + lane; tt < K_; tt += 32) out[tt] = first;
  }
}

// ---------------------------------------------------------------------------
// Transpose features [b][c][n] -> featT [b][n][c] so gathers are 512B bursts.
// ---------------------------------------------------------------------------
__global__ __launch_bounds__(256)
void transpose_feat(const float* __restrict__ f, float* __restrict__ ft) {
  __shared__ float tile[32][33];
  const int blk = blockIdx.x;
  const int b   = blk >> 9;             // 512 tiles per batch (4 * 128)
  const int rem = blk & 511;
  const int c0  = (rem >> 7) << 5;
  const int n0  = (rem & 127) << 5;
  const int tx  = threadIdx.x & 31;
  const int ty  = threadIdx.x >> 5;     // 0..7
  const float* fb = f + (size_t)b * C_ * N_;
  float* fob = ft + (size_t)b * N_ * C_;
  for (int i = 0; i < 4; ++i)
    tile[ty + i * 8][tx] = fb[(size_t)(c0 + ty + i * 8) * N_ + n0 + tx];
  __syncthreads();
  for (int i = 0; i < 4; ++i)
    fob[(size_t)(n0 + ty + i * 8) * C_ + c0 + tx] = tile[tx][ty + i * 8];
}

// ---------------------------------------------------------------------------
// Fused MLP layer GEMM on fp32 WMMA (V_WMMA_F32_16X16X4_F32).
//   LAYER 0: columns gathered from (xyz - new_xyz) ++ featT[idx]
//   LAYER 1/2: columns = relu(scale*prev + shift) applied on load
//   LAYER 2: fused max/min over the 32 samples instead of storing g2
// Deterministic BN stats: shfl-reduce -> per-wave LDS slots -> per-block
// partial record; finalize kernel folds into per-channel scale/shift.
// ---------------------------------------------------------------------------
template<int LAYER>
__global__ __launch_bounds__(128)
void mlp_gemm(const float* __restrict__ W, const float* __restrict__ bias,
              const float* __restrict__ xyz, const float* __restrict__ nxyz,
              const float* __restrict__ featT, const int* __restrict__ idx,
              const float* __restrict__ gIn, const float* __restrict__ scale,
              const float* __restrict__ shift,
              float* __restrict__ gOut, float* __restrict__ partial,
              float* __restrict__ gmax, float* __restrict__ gmin) {
  constexpr int KSTEPS = (LAYER == 0) ? 33 : 32;     // K padded to 132 for L0
  constexpr int KW     = (LAYER == 0) ? 131 : 128;   // weight row stride
  constexpr int COUT   = (LAYER == 2) ? 256 : 128;
  constexpr int MT     = COUT / 16;

  __shared__ float Xs[64][136];     // staged input columns (K-major per col)
  __shared__ float wS[4][256];      // per-wave channel sums
  __shared__ float wQ[4][256];      // per-wave channel sum-squares
  __shared__ float wMx[4][256];     // layer2: per-wave max over its 16 kk
  __shared__ float wMn[4][256];

  const int t     = threadIdx.x;
  const int pBase = blockIdx.x * 64;

  // ---- load/stage phase -------------------------------------------------
  {
    const int col  = t & 63;
    const int half = t >> 6;
    const int p    = pBase + col;
    if constexpr (LAYER == 0) {
      const int n  = idx[p];
      const int bs = p >> 5;
      const int b  = bs >> 10;
      const float4* fr4 =
          (const float4*)(featT + ((size_t)b * N_ + n) * C_ + half * 64);
      #pragma unroll
      for (int i = 0; i < 16; ++i) {
        float4 v = fr4[i];
        int c = 3 + half * 64 + i * 4;
        Xs[col][c + 0] = v.x; Xs[col][c + 1] = v.y;
        Xs[col][c + 2] = v.z; Xs[col][c + 3] = v.w;
      }
      if (half == 0) {
        const float* xp = xyz + ((size_t)b * N_ + n) * 3;
        Xs[col][0] = xp[0] - nxyz[bs * 3 + 0];
        Xs[col][1] = xp[1] - nxyz[bs * 3 + 1];
        Xs[col][2] = xp[2] - nxyz[bs * 3 + 2];
        Xs[col][131] = 0.0f;   // zero pad channel
      }
    } else {
      #pragma unroll 4
      for (int i = 0; i < 64; ++i) {
        int c = half * 64 + i;
        float v = gIn[(size_t)c * P_ + p];
        Xs[col][c] = fmaxf(0.0f, scale[c] * v + shift[c]);  // fused BN+ReLU
      }
    }
  }
  __syncthreads();

  // ---- compute phase ----------------------------------------------------
  const int wv   = t >> 5;
  const int lane = t & 31;
  const int hi   = lane >> 4;            // 0: K even pair, 1: K odd pair
  const int cw   = wv * 16 + (lane & 15); // this wave's column for B/D frags
  const int p    = pBase + cw;

  for (int mt = 0; mt < MT; ++mt) {
    const int m0 = mt * 16;
    v8f acc = {0.f, 0.f, 0.f, 0.f, 0.f, 0.f, 0.f, 0.f};
    const float* wr = W + (size_t)(m0 + (lane & 15)) * KW;
    #pragma unroll 4
    for (int ks = 0; ks < KSTEPS; ++ks) {
      const int k = ks * 4 + hi * 2;
      v2f a, bf;
      if constexpr (LAYER == 0) {
        a.x = (k     < 131) ? wr[k]     : 0.0f;
        a.y = (k + 1 < 131) ? wr[k + 1] : 0.0f;
      } else {
        a.x = wr[k];
        a.y = wr[k + 1];
      }
      bf.x = Xs[cw][k];
      bf.y = Xs[cw][k + 1];
      acc = __builtin_amdgcn_wmma_f32_16x16x4_f32(
          false, a, false, bf, (short)0, acc, false, false);
    }

    // ---- epilogue: bias, store / max-min, BN partial stats --------------
    #pragma unroll
    for (int v = 0; v < 8; ++v) {
      const int row = m0 + hi * 8 + v;
      float val = acc[v] + bias[row];
      if constexpr (LAYER != 2) gOut[(size_t)row * P_ + p] = val;
      float sv = val, qv = val * val, mx = val, mn = val;
      #pragma unroll
      for (int d = 1; d < 16; d <<= 1) {  // stays within 16-lane half
        sv += __shfl_xor(sv, d, 32);
        qv += __shfl_xor(qv, d, 32);
        if constexpr (LAYER == 2) {
          mx = fmaxf(mx, __shfl_xor(mx, d, 32));
          mn = fminf(mn, __shfl_xor(mn, d, 32));
        }
      }
      if ((lane & 15) == 0) {             // lanes 0 and 16, distinct rows
        wS[wv][row] = sv;
        wQ[wv][row] = qv;
        if constexpr (LAYER == 2) { wMx[wv][row] = mx; wMn[wv][row] = mn; }
      }
    }
  }
  __syncthreads();

  for (int c = t; c < COUT; c += 128) {
    float s = wS[0][c] + wS[1][c] + wS[2][c] + wS[3][c];
    float q = wQ[0][c] + wQ[1][c] + wQ[2][c] + wQ[3][c];
    partial[(size_t)blockIdx.x * 512 + c]       = s;
    partial[(size_t)blockIdx.x * 512 + 256 + c] = q;
  }
  if constexpr (LAYER == 2) {
    // waves {0,1} cover kk 0..31 of bs group 0, waves {2,3} of group 1
    for (int r = t; r < 512; r += 128) {
      int bl = r >> 8, row = r & 255;
      int bs = blockIdx.x * 2 + bl;
      gmax[(size_t)row * BS_ + bs] = fmaxf(wMx[2 * bl][row], wMx[2 * bl + 1][row]);
      gmin[(size_t)row * BS_ + bs] = fminf(wMn[2 * bl][row], wMn[2 * bl + 1][row]);
    }
  }
}

// ---------------------------------------------------------------------------
// Fold batch statistics into per-channel scale/shift (deterministic sum).
// ---------------------------------------------------------------------------
__global__ __launch_bounds__(256)
void finalize_bn(const float* __restrict__ partial,
                 const float* __restrict__ gamma, const float* __restrict__ beta,
                 float* __restrict__ scale, float* __restrict__ shift,
                 int cout, int nblk) {
  int c = threadIdx.x;
  if (c >= cout) return;
  float s = 0.f, q = 0.f;
  for (int b = 0; b < nblk; ++b) {
    s += partial[(size_t)b * 512 + c];
    q += partial[(size_t)b * 512 + 256 + c];
  }
  float mean = s * (1.0f / P_);
  float var  = q * (1.0f / P_) - mean * mean;
  float sc   = gamma[c] * rsqrtf(var + EPS_);
  scale[c] = sc;
  shift[c] = beta[c] - mean * sc;
}

// ---------------------------------------------------------------------------
// Final: new_features[b][c][s] = max over k of relu(sc*g2+sh)
//   = max(relu(sc*max_k g2 + sh), relu(sc*min_k g2 + sh))  (any sign of sc)
// ---------------------------------------------------------------------------
__global__ __launch_bounds__(256)
void final_out(const float* __restrict__ gmax, const float* __restrict__ gmin,
               const float* __restrict__ scale, const float* __restrict__ shift,
               float* __restrict__ out) {
  int t = blockIdx.x * 256 + threadIdx.x;   // (b*256 + c)*1024 + s
  if (t >= B_ * 256 * S_) return;
  int s = t & (S_ - 1);
  int c = (t >> 10) & 255;
  int b = t >> 18;
  int bs = b * S_ + s;
  float sc = scale[c], sh = shift[c];
  float vmax = fmaxf(0.0f, sc * gmax[(size_t)c * BS_ + bs] + sh);
  float vmin = fmaxf(0.0f, sc * gmin[(size_t)c * BS_ + bs] + sh);
  out[t] = fmaxf(vmax, vmin);
}

// ---------------------------------------------------------------------------
extern "C" void kernel_launch(void* const* d_in, const int* in_sizes, int n_in,
                              void* d_out, int out_size, void* d_ws, size_t ws_size,
                              hipStream_t stream) {
  (void)in_sizes; (void)n_in; (void)out_size; (void)ws_size;
  const float* xyz  = (const float*)d_in[0];
  const float* feat = (const float*)d_in[1];
  const float* w0 = (const float*)d_in[2];
  const float* b0 = (const float*)d_in[3];
  const float* ga0 = (const float*)d_in[4];
  const float* be0 = (const float*)d_in[5];
  const float* w1 = (const float*)d_in[6];
  const float* b1 = (const float*)d_in[7];
  const float* ga1 = (const float*)d_in[8];
  const float* be1 = (const float*)d_in[9];
  const float* w2 = (const float*)d_in[10];
  const float* b2 = (const float*)d_in[11];
  const float* ga2 = (const float*)d_in[12];
  const float* be2 = (const float*)d_in[13];

  float* out_xyz  = (float*)d_out;             // (B, S, 3)
  float* out_feat = out_xyz + (size_t)BS_ * 3; // (B, 256, S)

  char* ws = (char*)d_ws;
  size_t off = 0;
  auto alloc = [&](size_t bytes) {
    void* p = ws + off;
    off = (off + bytes + 255) & ~(size_t)255;
    return p;
  };
  int*   fps     = (int*)  alloc((size_t)BS_ * 4);
  int*   idx     = (int*)  alloc((size_t)P_ * 4);
  float* stats   = (float*)alloc((size_t)3 * 512 * 4);       // scale/shift x3
  float* featT   = (float*)alloc((size_t)B_ * N_ * C_ * 4);  // 32 MB
  float* partial = (float*)alloc((size_t)NBLK_ * 512 * 4);   // 16 MB
  float* g0buf   = (float*)alloc((size_t)C_ * P_ * 4);       // 256 MB
  float* g1buf   = (float*)alloc((size_t)C_ * P_ * 4);       // 256 MB
  float* gmaxb   = (float*)alloc((size_t)256 * BS_ * 4);     // 16 MB
  float* gminb   = (float*)alloc((size_t)256 * BS_ * 4);     // 16 MB

  fps_kernel<<<B_, 256, 0, stream>>>(xyz, fps);
  gather_newxyz<<<(BS_ + 255) / 256, 256, 0, stream>>>(xyz, fps, out_xyz);
  ball_query_kernel<<<BS_ / 4, 128, 0, stream>>>(xyz, out_xyz, idx);
  transpose_feat<<<B_ * 512, 256, 0, stream>>>(feat, featT);

  mlp_gemm<0><<<NBLK_, 128, 0, stream>>>(w0, b0, xyz, out_xyz, featT, idx,
                                         nullptr, nullptr, nullptr,
                                         g0buf, partial, nullptr, nullptr);
  finalize_bn<<<1, 256, 0, stream>>>(partial, ga0, be0,
                                     stats + 0, stats + 256, 128, NBLK_);

  mlp_gemm<1><<<NBLK_, 128, 0, stream>>>(w1, b1, xyz, out_xyz, featT, idx,
                                         g0buf, stats + 0, stats + 256,
                                         g1buf, partial, nullptr, nullptr);
  finalize_bn<<<1, 256, 0, stream>>>(partial, ga1, be1,
                                     stats + 512, stats + 768, 128, NBLK_);

  mlp_gemm<2><<<NBLK_, 128, 0, stream>>>(w2, b2, xyz, out_xyz, featT, idx,
                                         g1buf, stats + 512, stats + 768,
                                         nullptr, partial, gmaxb, gminb);
  finalize_bn<<<1, 256, 0, stream>>>(partial, ga2, be2,
                                     stats + 1024, stats + 1280, 256, NBLK_);

  final_out<<<(B_ * 256 * S_) / 256, 256, 0, stream>>>(gmaxb, gminb,
                                                       stats + 1024, stats + 1280,
                                                       out_feat);
}